// BPPGCN_36129264894354
// MI455X (gfx1250) — compile-verified
//
#include <hip/hip_runtime.h>

// CDNA5 / gfx1250: wave32, WMMA 16x16 tiles.
typedef float v2f __attribute__((ext_vector_type(2)));
typedef float v8f __attribute__((ext_vector_type(8)));

#define HDIM 64

// ---------------------------------------------------------------------------
// K2: deg[dst] += w  (self-loop weight 1 folded analytically later)
// ---------------------------------------------------------------------------
__global__ void gcn_deg_kernel(const int* __restrict__ dst,
                               const float* __restrict__ w,
                               float* __restrict__ deg, int E) {
    int e = blockIdx.x * blockDim.x + threadIdx.x;
    if (e < E) atomicAdd(&deg[dst[e]], w[e]);
}

// ---------------------------------------------------------------------------
// K3: deg -> dinv in place. deg+1 >= 1 always (self-loop), so no zero branch.
// ---------------------------------------------------------------------------
__global__ void gcn_dinv_kernel(float* __restrict__ deg, int N) {
    int i = blockIdx.x * blockDim.x + threadIdx.x;
    if (i < N) deg[i] = __frsqrt_rn(deg[i] + 1.0f);
}

// ---------------------------------------------------------------------------
// K4: norm[e] = dinv[src]*w*dinv[dst]; ssum[dst] += norm  (layer-1 scalar)
// ---------------------------------------------------------------------------
__global__ void gcn_norm_kernel(const int* __restrict__ src,
                                const int* __restrict__ dst,
                                const float* __restrict__ w,
                                const float* __restrict__ dinv,
                                float* __restrict__ nrm,
                                float* __restrict__ ssum, int E) {
    int e = blockIdx.x * blockDim.x + threadIdx.x;
    if (e < E) {
        int d = dst[e];
        float v = dinv[src[e]] * w[e] * dinv[d];
        nrm[e] = v;
        atomicAdd(&ssum[d], v);
    }
}

// ---------------------------------------------------------------------------
// K5: fused layer-1 + GEMM:  h2 = relu(s_tot * W1 + b1) @ W2
// One wave32 computes a 16-row x 64-col tile with V_WMMA_F32_16X16X4_F32
// (pure fp32 tensor path -> bit-compatible math with the fp32 reference).
//
// A (16x4 f32): lanes 0-15 rows M=0..15 holding K={k0,k0+1};
//               lanes 16-31 same rows holding K={k0+2,k0+3}.   (ISA 7.12.2)
// B (4x16 f32): lanes 0-15 cols N=0..15 holding K={k0,k0+1};
//               lanes 16-31 same cols holding K={k0+2,k0+3}.
// C/D (16x16 f32, 8 VGPRs): VGPR r = row r (lanes 0-15) / row 8+r (lanes 16-31).
// ---------------------------------------------------------------------------
__global__ void gcn_h2_wmma_kernel(const float* __restrict__ ssum,
                                   const float* __restrict__ dinv,
                                   const float* __restrict__ W1,
                                   const float* __restrict__ b1,
                                   const float* __restrict__ W2,
                                   float* __restrict__ h2, int N) {
    const int lane = threadIdx.x & 31;
    const int wave = threadIdx.x >> 5;
    const int wavesPerBlock = blockDim.x >> 5;
    const int tile = blockIdx.x * wavesPerBlock + wave;   // wave-uniform
    const int row0 = tile * 16;
    if (row0 >= N) return;                                // wave-uniform exit

    const int half = lane >> 4;     // 0: lanes 0-15, 1: lanes 16-31
    const int l15  = lane & 15;

    // Per-lane row (both halves cover the same 16 rows).
    const int m = row0 + l15;
    float sv = 0.0f, dv = 0.0f;
    if (m < N) { sv = ssum[m]; dv = dinv[m]; }
    const float stot = sv + dv * dv;                      // + self-loop norm

    v8f c0 = {}, c1 = {}, c2 = {}, c3 = {};

    #pragma unroll
    for (int k0 = 0; k0 < HDIM; k0 += 4) {
        const int ka = k0 + 2 * half;
        // A fragment: x1[m, ka..ka+1] = relu(stot*W1 + b1), built in-registers.
        v2f a;
        a.x = fmaxf(stot * W1[ka]     + b1[ka],     0.0f);
        a.y = fmaxf(stot * W1[ka + 1] + b1[ka + 1], 0.0f);

        // B fragments for the 4 output column tiles (n0 = 0,16,32,48).
        const float* r0 = W2 + (size_t)ka * HDIM + l15;
        const float* r1 = r0 + HDIM;
        v2f b0; b0.x = r0[0];  b0.y = r1[0];
        v2f b1v; b1v.x = r0[16]; b1v.y = r1[16];
        v2f b2v; b2v.x = r0[32]; b2v.y = r1[32];
        v2f b3v; b3v.x = r0[48]; b3v.y = r1[48];

        c0 = __builtin_amdgcn_wmma_f32_16x16x4_f32(false, a, false, b0,  (short)0, c0, false, false);
        c1 = __builtin_amdgcn_wmma_f32_16x16x4_f32(false, a, false, b1v, (short)0, c1, false, false);
        c2 = __builtin_amdgcn_wmma_f32_16x16x4_f32(false, a, false, b2v, (short)0, c2, false, false);
        c3 = __builtin_amdgcn_wmma_f32_16x16x4_f32(false, a, false, b3v, (short)0, c3, false, false);
    }

    // Store D tiles: row = row0 + r + 8*half, col = n0 + l15.
    #pragma unroll
    for (int r = 0; r < 8; ++r) {
        const int row = row0 + r + 8 * half;
        if (row < N) {
            float* o = h2 + (size_t)row * HDIM + l15;
            o[0]  = c0[r];
            o[16] = c1[r];
            o[32] = c2[r];
            o[48] = c3[r];
        }
    }
}

// ---------------------------------------------------------------------------
// K6: layer-2 propagation. 16 lanes per edge, float4 (128-bit) gathers from
// h2 (L2-resident, 25.6MB), 4 fp32 atomics each into out (L2-resident).
// ---------------------------------------------------------------------------
__global__ void gcn_scatter_kernel(const int* __restrict__ src,
                                   const int* __restrict__ dst,
                                   const float* __restrict__ nrm,
                                   const float* __restrict__ h2,
                                   float* __restrict__ out, int E) {
    const long long gid = (long long)blockIdx.x * blockDim.x + threadIdx.x;
    const long long e = gid >> 4;
    if (e >= E) return;
    const int seg = (int)(gid & 15);

    const int   s = src[e];
    const int   d = dst[e];
    const float v = nrm[e];

    const float4 hv = *(const float4*)(h2 + (size_t)s * HDIM + seg * 4);
    float* o = out + (size_t)d * HDIM + seg * 4;
    atomicAdd(o + 0, v * hv.x);
    atomicAdd(o + 1, v * hv.y);
    atomicAdd(o + 2, v * hv.z);
    atomicAdd(o + 3, v * hv.w);
}

// ---------------------------------------------------------------------------
// K7: epilogue: fold the self-loop edge (norm = dinv^2, src = dst = i) here
// instead of N*64 extra atomics, add bias, relu.
// ---------------------------------------------------------------------------
__global__ void gcn_final_kernel(const float* __restrict__ dinv,
                                 const float* __restrict__ h2,
                                 const float* __restrict__ b2,
                                 float* __restrict__ out, int N) {
    const int gid = blockIdx.x * blockDim.x + threadIdx.x;
    if (gid < N * HDIM) {
        const int i = gid >> 6;
        const int h = gid & (HDIM - 1);
        const float di = dinv[i];
        out[gid] = fmaxf(out[gid] + di * di * h2[gid] + b2[h], 0.0f);
    }
}

// ---------------------------------------------------------------------------
extern "C" void kernel_launch(void* const* d_in, const int* in_sizes, int n_in,
                              void* d_out, int out_size, void* d_ws, size_t ws_size,
                              hipStream_t stream) {
    const int*   edge_index = (const int*)d_in[0];    // [2, E] int32
    const float* edge_attr  = (const float*)d_in[1];  // [E]
    // d_in[2] = num_nodes (device scalar; N derived host-side instead)
    const float* W1 = (const float*)d_in[3];          // [1, 64]
    const float* b1 = (const float*)d_in[4];          // [64]
    const float* W2 = (const float*)d_in[5];          // [64, 64] (k, n)
    const float* b2 = (const float*)d_in[6];          // [64]

    const int E = in_sizes[1];
    const int N = out_size / HDIM;
    const int* src = edge_index;
    const int* dst = edge_index + E;
    float* out = (float*)d_out;

    // Workspace: deg/dinv [N], ssum [N], norm [E], h2 [N*64]  (~39.2 MB)
    float* deg  = (float*)d_ws;
    float* ssum = deg + N;
    float* nrm  = ssum + N;
    float* h2   = nrm + E;

    hipMemsetAsync(deg,  0, (size_t)N * sizeof(float), stream);
    hipMemsetAsync(ssum, 0, (size_t)N * sizeof(float), stream);
    hipMemsetAsync(out,  0, (size_t)N * HDIM * sizeof(float), stream);

    const int T = 256;
    gcn_deg_kernel <<<(E + T - 1) / T, T, 0, stream>>>(dst, edge_attr, deg, E);
    gcn_dinv_kernel<<<(N + T - 1) / T, T, 0, stream>>>(deg, N);  // deg -> dinv
    gcn_norm_kernel<<<(E + T - 1) / T, T, 0, stream>>>(src, dst, edge_attr, deg, nrm, ssum, E);

    const int tiles = (N + 15) / 16;          // one wave32 per 16-row tile
    const int wavesPerBlock = T / 32;
    gcn_h2_wmma_kernel<<<(tiles + wavesPerBlock - 1) / wavesPerBlock, T, 0, stream>>>(
        ssum, deg, W1, b1, W2, h2, N);

    const long long threads = (long long)E * 16;
    gcn_scatter_kernel<<<(unsigned)((threads + T - 1) / T), T, 0, stream>>>(
        src, dst, nrm, h2, out, E);

    gcn_final_kernel<<<(N * HDIM + T - 1) / T, T, 0, stream>>>(deg, h2, b2, out, N);
}